// VectorQuantizerEMA_37649683317552
// MI455X (gfx1250) — compile-verified
//
#include <hip/hip_runtime.h>
#include <hip/hip_bf16.h>
#include <math.h>

// ---------------------------------------------------------------- constants
#define N_ROWS 32768   // 16 * 2048
#define DIM    64
#define KCODES 1024
#define MT     128     // rows per workgroup in main kernel
#define NWAVES 8       // 256 threads

#define LOG2E 1.4426950408889634f
#define LN2   0.6931471805599453f

// workspace layout (floats)
#define WS_COMMIT 0
#define WS_ENT    1
#define WS_COS2   2
#define WS_NUSED  3
#define WS_HIST   4                    // KCODES entries
#define WS_NORMED (4 + KCODES)         // KCODES*DIM entries

typedef float v2f __attribute__((ext_vector_type(2)));
typedef float v8f __attribute__((ext_vector_type(8)));
typedef int   v4i __attribute__((ext_vector_type(4)));

// ---------------------------------------------------------------- fast math
__device__ __forceinline__ float fexp2(float x) {
#if __has_builtin(__builtin_amdgcn_exp2f)
  return __builtin_amdgcn_exp2f(x);      // raw v_exp_f32
#else
  return exp2f(x);
#endif
}
__device__ __forceinline__ float flog2(float x) {
#if __has_builtin(__builtin_amdgcn_logf)
  return __builtin_amdgcn_logf(x);       // raw v_log_f32
#else
  return log2f(x);
#endif
}

// ---------------------------------------------------------------- async copy
#if __has_builtin(__builtin_amdgcn_global_load_async_to_lds_b128)
#define HAVE_ASYNC_LDS 1
#else
#define HAVE_ASYNC_LDS 0
#endif

__device__ __forceinline__ void stage_b128(void* dst_lds, const void* src_g) {
#if HAVE_ASYNC_LDS
  __builtin_amdgcn_global_load_async_to_lds_b128(
      (__attribute__((address_space(1))) v4i*)src_g,
      (__attribute__((address_space(3))) v4i*)dst_lds, 0, 0);
#else
  *(float4*)dst_lds = *(const float4*)src_g;
#endif
}

__device__ __forceinline__ void wait_async_all() {
#if HAVE_ASYNC_LDS
#if __has_builtin(__builtin_amdgcn_s_wait_asynccnt)
  __builtin_amdgcn_s_wait_asynccnt(0);
#else
  asm volatile("s_wait_asynccnt 0" ::: "memory");
#endif
#endif
}

// ---------------------------------------------------------------- init ws
__global__ void vq_init(float* __restrict__ ws) {
  int i = blockIdx.x * blockDim.x + threadIdx.x;
  if (i < 4 + KCODES) ws[i] = 0.0f;
}

// ---------------------------------------------------------------- normalize codebook
__global__ void vq_norm(const float* __restrict__ E,
                        const float* __restrict__ code_count,
                        float* __restrict__ ws) {
  __shared__ float red[DIM];
  const int k = blockIdx.x;
  const int t = threadIdx.x;                 // 64 threads
  float v = E[k * DIM + t];
  red[t] = v * v;
  __syncthreads();
  for (int o = DIM / 2; o > 0; o >>= 1) {
    if (t < o) red[t] += red[t + o];
    __syncthreads();
  }
  float nrm  = sqrtf(red[0]);
  float mask = (code_count[k] >= 1.0f) ? 1.0f : 0.0f;
  float inv  = mask / fmaxf(nrm, 1e-12f);
  ws[WS_NORMED + k * DIM + t] = v * inv;
  if (t == 0) atomicAdd(&ws[WS_NUSED], mask);
}

// ---------------------------------------------------------------- main fused kernel
// 256 threads = 8 waves. Block handles MT=128 rows; wave w handles rows
// [16w, 16w+16). Online softmax state is LANE-LOCAL (each lane sees a
// disjoint 64-column subset of each row); lanes merge once at the end.
__global__ __launch_bounds__(256, 1)
void vq_main(const float* __restrict__ X, const float* __restrict__ E,
             float* __restrict__ qout, float* __restrict__ idxout,
             float* __restrict__ ws) {
  __shared__ float sE[KCODES * DIM];   // 256 KB: full codebook
  __shared__ float sX[MT * DIM];       //  32 KB: row tile
  __shared__ float sE2[KCODES];        //   4 KB: ||e_k||^2 * log2(e)
  __shared__ float sXn2[MT];           // ||x_n||^2

  const int tid  = threadIdx.x;
  const int wave = tid >> 5;
  const int lane = tid & 31;
  const int row0 = blockIdx.x * MT;

  // ---- stage codebook + X tile into LDS (async DMA path on gfx1250)
  {
    const float4* gE = (const float4*)E;
    float4*       lE = (float4*)sE;
    for (int i = tid; i < KCODES * DIM / 4; i += 256) stage_b128(&lE[i], &gE[i]);
    const float4* gX = (const float4*)(X + (size_t)row0 * DIM);
    float4*       lX = (float4*)sX;
    for (int i = tid; i < MT * DIM / 4; i += 256) stage_b128(&lX[i], &gX[i]);
    wait_async_all();
  }
  __syncthreads();

  // ---- squared norms (codebook norms pre-scaled into exp2 domain)
  for (int k = tid; k < KCODES; k += 256) {
    const float* e = &sE[k * DIM];
    float s = 0.0f;
#pragma unroll
    for (int d = 0; d < DIM; ++d) s += e[d] * e[d];
    sE2[k] = s * LOG2E;
  }
  if (tid < MT) {
    const float* x = &sX[tid * DIM];
    float s = 0.0f;
#pragma unroll
    for (int d = 0; d < DIM; ++d) s += x[d] * x[d];
    sXn2[tid] = s;
  }
  __syncthreads();

  const int mrow = wave * 16;          // wave's row-tile base (local)
  const int r    = lane & 15;
  const int hi   = lane >> 4;          // 0: rows i..i, 1: rows i+8

  // lane-local online softmax state per C-VGPR row group (exp2 domain):
  // m = max z2 (doubles as argmax value), S = sum 2^(z2-m), W = sum z2*2^(z2-m)
  float m[8], S[8], W[8];
  int   bidx[8];
#pragma unroll
  for (int i = 0; i < 8; ++i) { m[i] = -3.0e38f; S[i] = 0.0f; W[i] = 0.0f; bidx[i] = 0; }

  for (int ct = 0; ct < KCODES / 16; ++ct) {
    const int cbase = ct * 16;
    // two independent WMMA accumulation chains for ILP
    v8f acc0 = {}, acc1 = {};
#pragma unroll
    for (int kk = 0; kk < 8; ++kk) {
      const int kb0 = (2 * kk) * 4 + hi * 2;
      const int kb1 = kb0 + 4;
      v2f a0 = *(const v2f*)(sX + (mrow + r) * DIM + kb0);
      v2f b0 = *(const v2f*)(sE + (cbase + r) * DIM + kb0);
      acc0 = __builtin_amdgcn_wmma_f32_16x16x4_f32(false, a0, false, b0,
                                                   (short)0, acc0, false, false);
      v2f a1 = *(const v2f*)(sX + (mrow + r) * DIM + kb1);
      v2f b1 = *(const v2f*)(sE + (cbase + r) * DIM + kb1);
      acc1 = __builtin_amdgcn_wmma_f32_16x16x4_f32(false, a1, false, b1,
                                                   (short)0, acc1, false, false);
    }
    const float e2c = sE2[cbase + r];   // ||e_col||^2 * log2e
    const int   col = cbase + r;

#pragma unroll
    for (int i = 0; i < 8; ++i) {
      // z2 = (2*x.e - ||e||^2) * log2e   (softmax logit, exp2 domain)
      float z2 = __builtin_fmaf(acc0[i] + acc1[i], 2.0f * LOG2E, -e2c);
      float mo = m[i];
      bool  gt = z2 > mo;                    // one compare feeds both selects
      float nm = gt ? z2 : mo;
      bidx[i]  = gt ? col : bidx[i];
      float e1 = fexp2(mo - nm);
      float ev = fexp2(z2 - nm);
      S[i] = __builtin_fmaf(S[i], e1, ev);
      W[i] = __builtin_fmaf(W[i], e1, z2 * ev);
      m[i] = nm;
    }
  }

  // ---- merge lane-local states across the 16 lanes of each row (once)
#pragma unroll
  for (int i = 0; i < 8; ++i) {
#pragma unroll
    for (int o = 1; o < 16; o <<= 1) {
      float om = __shfl_xor(m[i], o, 32);
      float oS = __shfl_xor(S[i], o, 32);
      float oW = __shfl_xor(W[i], o, 32);
      int   oi = __shfl_xor(bidx[i], o, 32);
      bool  gt = om > m[i];
      float nm = gt ? om : m[i];
      bidx[i]  = gt ? oi : bidx[i];
      float e1 = fexp2(m[i] - nm);
      float e2v = fexp2(om - nm);
      S[i] = S[i] * e1 + oS * e2v;
      W[i] = W[i] * e1 + oW * e2v;
      m[i] = nm;
    }
  }

  // ---- per-row outputs
#pragma unroll
  for (int i = 0; i < 8; ++i) {
    const int rl   = mrow + i + 8 * hi;            // local row in block tile
    const int grow = row0 + rl;                    // global row
    const int k    = bidx[i];
    if (r == 0) {
      // back to natural log domain: z_max = m*ln2, lnS = ln2*log2(S),
      // W_nat = ln2 * W ; entropy = z_max + lnS - W_nat/S
      float zmax = m[i] * LN2;
      float ent  = zmax + LN2 * flog2(S[i]) - LN2 * (W[i] / S[i]);
      float dmin = sXn2[rl] - zmax;                // ||x - e_k||^2
      atomicAdd(&ws[WS_COMMIT], dmin);
      atomicAdd(&ws[WS_ENT], ent);
      atomicAdd(&ws[WS_HIST + k], 1.0f);
      idxout[grow] = (float)k;
    }
    // 16 lanes of this half write the quantized row (4 floats each)
    float4 q = ((const float4*)&sE[k * DIM])[r];
    ((float4*)(qout + (size_t)grow * DIM))[r] = q;
  }
}

// ---------------------------------------------------------------- ortho loss (WMMA)
// cos = normed @ normed^T ; accumulate sum(cos^2). 64x64 tiles of 16x16,
// one wave per tile, 8 waves per block -> 512 blocks.
__global__ __launch_bounds__(256, 1)
void vq_ortho(float* __restrict__ ws) {
  const float* Nd = ws + WS_NORMED;
  const int wave = threadIdx.x >> 5;
  const int lane = threadIdx.x & 31;
  const int tile = blockIdx.x * NWAVES + wave;
  const int ti = tile >> 6, tj = tile & 63;
  const int r = lane & 15, hi = lane >> 4;
  const int ibase = ti * 16, jbase = tj * 16;

  v8f acc0 = {}, acc1 = {};
#pragma unroll
  for (int kk = 0; kk < 8; ++kk) {
    const int kb0 = (2 * kk) * 4 + hi * 2;
    const int kb1 = kb0 + 4;
    v2f a0 = *(const v2f*)(Nd + (ibase + r) * DIM + kb0);
    v2f b0 = *(const v2f*)(Nd + (jbase + r) * DIM + kb0);
    acc0 = __builtin_amdgcn_wmma_f32_16x16x4_f32(false, a0, false, b0,
                                                 (short)0, acc0, false, false);
    v2f a1 = *(const v2f*)(Nd + (ibase + r) * DIM + kb1);
    v2f b1 = *(const v2f*)(Nd + (jbase + r) * DIM + kb1);
    acc1 = __builtin_amdgcn_wmma_f32_16x16x4_f32(false, a1, false, b1,
                                                 (short)0, acc1, false, false);
  }
  float s = 0.0f;
#pragma unroll
  for (int i = 0; i < 8; ++i) {
    float c = acc0[i] + acc1[i];
    s += c * c;
  }
#pragma unroll
  for (int o = 1; o < 32; o <<= 1) s += __shfl_xor(s, o, 32);
  if (lane == 0) atomicAdd(&ws[WS_COS2], s);
}

// ---------------------------------------------------------------- finalize scalars
__global__ void vq_final(const float* __restrict__ ws, float* __restrict__ scal) {
  __shared__ float red[KCODES];
  const int t = threadIdx.x;                       // KCODES threads
  float h  = ws[WS_HIST + t] * (1.0f / (float)N_ROWS);
  red[t] = h * __logf(h + 1e-10f);
  __syncthreads();
  for (int o = KCODES / 2; o > 0; o >>= 1) {
    if (t < o) red[t] += red[t + o];
    __syncthreads();
  }
  if (t == 0) {
    float nused = ws[WS_NUSED];
    scal[0] = ws[WS_COMMIT] / ((float)N_ROWS * (float)DIM);        // commitment
    scal[1] = ws[WS_COS2] / (nused * nused) - 1.0f / nused;        // ortho
    scal[2] = (ws[WS_ENT] / (float)N_ROWS) /
              (__logf((float)KCODES) / __logf(2.0f));              // entropy
    scal[3] = __expf(-red[0]);                                     // perplexity
    scal[4] = nused / (float)KCODES;                               // coverage
  }
}

// ---------------------------------------------------------------- launch
extern "C" void kernel_launch(void* const* d_in, const int* in_sizes, int n_in,
                              void* d_out, int out_size, void* d_ws, size_t ws_size,
                              hipStream_t stream) {
  (void)in_sizes; (void)n_in; (void)out_size; (void)ws_size;
  const float* X  = (const float*)d_in[0];   // inputs  [16,2048,64]
  const float* E  = (const float*)d_in[1];   // embedding [1024,64]
  const float* cc = (const float*)d_in[2];   // code_count [1024]

  float* out    = (float*)d_out;
  float* qout   = out;                                   // [N_ROWS*DIM]
  float* scal   = out + (size_t)N_ROWS * DIM;            // 5 scalars
  float* idxout = scal + 5;                              // [N_ROWS]
  float* ws     = (float*)d_ws;

  vq_init <<<(4 + KCODES + 255) / 256, 256, 0, stream>>>(ws);
  vq_norm <<<KCODES, DIM, 0, stream>>>(E, cc, ws);
  vq_main <<<N_ROWS / MT, 256, 0, stream>>>(X, E, qout, idxout, ws);
  vq_ortho<<<(KCODES / 16) * (KCODES / 16) / NWAVES, 256, 0, stream>>>(ws);
  vq_final<<<1, KCODES, 0, stream>>>(ws, scal);
}